// DropStripes_154618823374
// MI455X (gfx1250) — compile-verified
//
#include <hip/hip_runtime.h>

// DropStripes for MI455X (gfx1250, wave32).
// Memory-bound: ~1 GB traffic -> ~44 us roofline @ 23.3 TB/s. No matrix ops
// exist in this workload (1 select/multiply per element), so WMMA has no role.
// CDNA5-specific path: async global<->LDS DMA (ASYNCcnt) for the copy rows,
// avoiding the VGPR round-trip; dropped rows are write-only zero stores.

typedef float v4f __attribute__((ext_vector_type(4)));
typedef int   v4i __attribute__((ext_vector_type(4)));

constexpr int kB = 64;        // batch
constexpr int kC = 8;         // channels
constexpr int kT = 2048;      // time
constexpr int kF = 128;       // features
constexpr int kRows = 4;      // t-rows per block
constexpr int kThreads = 256; // 8 waves; wave w handles channel chunk c=w

#define AS_GLOBAL __attribute__((address_space(1)))
#define AS_LOCAL  __attribute__((address_space(3)))

#if defined(__AMDGCN__) &&                                              \
    __has_builtin(__builtin_amdgcn_global_load_async_to_lds_b128) &&    \
    __has_builtin(__builtin_amdgcn_global_store_async_from_lds_b128) && \
    __has_builtin(__builtin_amdgcn_s_wait_asynccnt)
#define USE_ASYNC_LDS 1
#else
#define USE_ASYNC_LDS 0
#endif

__global__ __launch_bounds__(kThreads) void drop_stripes_kernel(
    const float* __restrict__ in,
    const int*  __restrict__ distances,   // [B,2]
    const int*  __restrict__ bgns,        // [B,2]
    float* __restrict__ out)
{
#if USE_ASYNC_LDS
  __shared__ float tile[kRows * kC * kF];   // 16 KB staging (of 320 KB/WGP)
#endif

  const unsigned bt0 = blockIdx.x * kRows;  // first (b,t) row for this block
  const unsigned b   = bt0 >> 11;           // / kT (kT = 2048)
  const unsigned t0  = bt0 & (kT - 1);      // % kT (kRows divides kT)

  // Block-uniform -> scalar loads (2x s_load_b64)
  const int lo0 = bgns[2 * b + 0];
  const int lo1 = bgns[2 * b + 1];
  const int hi0 = lo0 + distances[2 * b + 0];
  const int hi1 = lo1 + distances[2 * b + 1];

  const unsigned i  = threadIdx.x;
  const unsigned c  = i >> 5;               // wave id = channel chunk
  const unsigned f4 = (i & 31u) << 2;       // float4 slot within F=128
  // Element offset of (b, c, t0, f4); tensor is 2^27 elements -> 32-bit safe.
  const unsigned base = ((b * kC + c) * kT + t0) * kF + f4;

  bool drop[kRows];
#pragma unroll
  for (int r = 0; r < kRows; ++r) {
    const int t = (int)t0 + r;
    drop[r] = ((t >= lo0) && (t < hi0)) || ((t >= lo1) && (t < hi1));
  }

#if USE_ASYNC_LDS
  // Phase 1: kept rows -> async DMA global->LDS (ASYNCcnt); each thread owns
  // a disjoint 16 B LDS slot, so per-wave ASYNCcnt ordering is sufficient.
#pragma unroll
  for (int r = 0; r < kRows; ++r) {
    if (!drop[r]) {   // uniform branch
      __builtin_amdgcn_global_load_async_to_lds_b128(
          (AS_GLOBAL v4i*)(in + base + (unsigned)(r * kF)),
          (AS_LOCAL  v4i*)(tile + r * (kC * kF) + (i << 2)),
          0, 0);
    }
  }
  // Dropped rows: write-only zero stores (skip the 512 B read entirely).
#pragma unroll
  for (int r = 0; r < kRows; ++r) {
    if (drop[r]) {
      v4f z = {0.f, 0.f, 0.f, 0.f};
      __builtin_nontemporal_store(z, (v4f*)(out + base + (unsigned)(r * kF)));
    }
  }
  __builtin_amdgcn_s_wait_asynccnt(0);   // our LDS slots are filled

  // Phase 2: kept rows -> async DMA LDS->global.
#pragma unroll
  for (int r = 0; r < kRows; ++r) {
    if (!drop[r]) {
      __builtin_amdgcn_global_store_async_from_lds_b128(
          (AS_GLOBAL v4i*)(out + base + (unsigned)(r * kF)),
          (AS_LOCAL  v4i*)(tile + r * (kC * kF) + (i << 2)),
          0, 0);
    }
  }
  __builtin_amdgcn_s_wait_asynccnt(0);   // drain (S_ENDPGM also waits idle)
#else
  // Fallback: VGPR-staged B128 NT streaming (proven codegen from round 1).
#pragma unroll
  for (int r = 0; r < kRows; ++r) {
    const unsigned off = base + (unsigned)(r * kF);
    if (drop[r]) {
      v4f z = {0.f, 0.f, 0.f, 0.f};
      __builtin_nontemporal_store(z, (v4f*)(out + off));
    } else {
      v4f v = __builtin_nontemporal_load((const v4f*)(in + off));
      __builtin_nontemporal_store(v, (v4f*)(out + off));
    }
  }
#endif
}

extern "C" void kernel_launch(void* const* d_in, const int* in_sizes, int n_in,
                              void* d_out, int out_size, void* d_ws, size_t ws_size,
                              hipStream_t stream) {
  const float* in        = (const float*)d_in[0];
  const int*   distances = (const int*)d_in[1];
  const int*   bgns      = (const int*)d_in[2];
  float*       out       = (float*)d_out;

  const int nBlocks = (kB * kT) / kRows;   // 32768 blocks x 256 threads
  drop_stripes_kernel<<<dim3(nBlocks), dim3(kThreads), 0, stream>>>(
      in, distances, bgns, out);
}